// TemporalSAGE_18313740550692
// MI455X (gfx1250) — compile-verified
//
#include <hip/hip_runtime.h>

// Problem constants from the reference
constexpr int NN  = 512;   // nodes
constexpr int DIN = 64;    // input dim
constexpr int DU  = 128;   // embedding dim
constexpr int DK  = 32;    // rel feature dim
constexpr int DE  = 288;   // 2*DU + DK

typedef float v2f __attribute__((ext_vector_type(2)));
typedef float v8f __attribute__((ext_vector_type(8)));

__device__ __forceinline__ float leaky(float x) { return x >= 0.f ? x : 0.01f * x; }

// ---------------------------------------------------------------------------
// seq0 = leaky(input @ W_seq + b_seq)   (512x64)@(64x128)
// ---------------------------------------------------------------------------
__global__ __launch_bounds__(256) void seq_embed_kernel(
    const float* __restrict__ in, const float* __restrict__ W,
    const float* __restrict__ b, float* __restrict__ seq0)
{
  int idx = blockIdx.x * 256 + threadIdx.x;          // 512*128
  int n = idx >> 7, u = idx & 127;
  float a = b[u];
  const float* row = in + n * DIN;
  #pragma unroll 4
  for (int k = 0; k < DIN; ++k) a += row[k] * W[k * DU + u];
  seq0[idx] = leaky(a);
}

// ---------------------------------------------------------------------------
// mask[i,j] = (sum_k rel[i,j,k] > 0);  deg[i] = sum_j mask[i,j]
// ---------------------------------------------------------------------------
__global__ __launch_bounds__(128) void mask_deg_kernel(
    const float* __restrict__ rel, float* __restrict__ mask, float* __restrict__ deg)
{
  __shared__ float red[128];
  int i = blockIdx.x, t = threadIdx.x;
  float cnt = 0.f;
  for (int j = t; j < NN; j += 128) {
    const float* p = rel + ((size_t)i * NN + j) * DK;
    float s = 0.f;
    #pragma unroll 8
    for (int k = 0; k < DK; ++k) s += p[k];
    float m = (s > 0.f) ? 1.f : 0.f;
    mask[(size_t)i * NN + j] = m;
    cnt += m;
  }
  red[t] = cnt; __syncthreads();
  for (int s = 64; s > 0; s >>= 1) { if (t < s) red[t] += red[t + s]; __syncthreads(); }
  if (t == 0) deg[i] = red[0];
}

// ---------------------------------------------------------------------------
// S1 = seq @ W1[0:128,:]   S2 = seq @ W1[128:256,:]   both (512x288)
// ---------------------------------------------------------------------------
__global__ __launch_bounds__(256) void s12_kernel(
    const float* __restrict__ seq, const float* __restrict__ W1,
    float* __restrict__ S1, float* __restrict__ S2)
{
  int idx = blockIdx.x * 256 + threadIdx.x;          // 512*288
  if (idx >= NN * DE) return;
  int n = idx / DE, e = idx - n * DE;
  const float* sr = seq + n * DU;
  float a = 0.f, b = 0.f;
  #pragma unroll 4
  for (int u = 0; u < DU; ++u) {
    float x = sr[u];
    a += x * W1[u * DE + e];
    b += x * W1[(DU + u) * DE + e];
  }
  S1[idx] = a; S2[idx] = b;
}

// ---------------------------------------------------------------------------
// Fused per-edge hop kernel using V_WMMA_F32_16X16X4_F32.
// Grid: (NN/16 j-blocks, 8 i-chunks), block = 128 (4 wave32; each wave: 16 i's).
// W1_rel is staged in LDS pre-blocked in WMMA-B lane order (one ds_load_b64
// per lane per k-step); the S2 j-tile (contiguous 18KB) is staged with the
// gfx1250 async LDS-fill path (global_load_async_to_lds_b128 + ASYNCcnt).
// Per edge-tile (16 edges x K=32):  C = rel_tile @ W1_rel, then in-register
// epilogue: pre = mask*(S1[i]+S2[j]+C)+b1; h=leaky(pre); dot += h*W2;
// cross-lane reduce -> w = leaky(dot+b2)*mask/deg[i] -> deterministic
// per-wave partials (no float atomics).
// ---------------------------------------------------------------------------
__global__ __launch_bounds__(128) void edge_hop_kernel(
    const float* __restrict__ rel, const float* __restrict__ W1rel,
    const float* __restrict__ b1, const float* __restrict__ W2,
    const float* __restrict__ b2, const float* __restrict__ S1,
    const float* __restrict__ S2, const float* __restrict__ mask,
    const float* __restrict__ deg, float* __restrict__ partials)
{
  // W1_rel blocked for WMMA-B: [(kb*18+nb)*32 + lane] -> v2f {row k0, row k0+1}
  __shared__ __align__(16) float ldsWB[8 * 18 * 32 * 2];   // 9216 f = 36 KB
  __shared__ __align__(16) float ldsS2[16 * DE];           // 4608 f = 18 KB
  __shared__ float ldsB1[DE];
  __shared__ float ldsW2[DE];

  const int j0  = blockIdx.x * 16;
  const int tid = threadIdx.x;

  // --- async DMA: S2 j-tile (contiguous) -> LDS, tracked by ASYNCcnt ---
  {
    const float* s2base = S2 + (size_t)j0 * DE;            // 16*288 floats
    for (int t = tid * 4; t < 16 * DE; t += 128 * 4) {
      unsigned ldsAddr = (unsigned)(size_t)&ldsS2[t];
      const float* g = s2base + t;
      asm volatile("global_load_async_to_lds_b128 %0, %1, off"
                   :: "v"(ldsAddr), "v"(g) : "memory");
    }
  }

  // --- stage W1_rel into WMMA-B blocked layout (regular loads, once/WG) ---
  for (int p = tid; p < 8 * 18 * 32; p += 128) {
    int kb  = p / (18 * 32);
    int rem = p - kb * (18 * 32);
    int nb  = rem >> 5;
    int l   = rem & 31;
    int k0  = kb * 4 + (l >> 4) * 2;
    int n   = nb * 16 + (l & 15);
    ldsWB[p * 2 + 0] = W1rel[(k0 + 0) * DE + n];
    ldsWB[p * 2 + 1] = W1rel[(k0 + 1) * DE + n];
  }
  for (int t = tid; t < DE; t += 128) { ldsB1[t] = b1[t]; ldsW2[t] = W2[t]; }

  asm volatile("s_wait_asynccnt 0x0" ::: "memory");
  __syncthreads();

  const int wave  = tid >> 5;
  const int lane  = tid & 31;
  const int half  = lane >> 4;       // K-split half (A: K {0,1}|{2,3}; C: M r | r+8)
  const int ln    = lane & 15;
  const int iBase = blockIdx.y * 64 + wave * 16;
  const float b2v = b2[0];

  float sacc = 0.f;                  // sum_i w[i, j_owned]

  for (int ii = 0; ii < 16; ++ii) {
    const int i = iBase + ii;
    const float* relRow = rel + ((size_t)i * NN + j0 + ln) * DK;
    if (ii + 1 < 16)
      __builtin_prefetch(rel + ((size_t)(i + 1) * NN + j0 + ln) * DK, 0, 0);

    // A tiles: 16x4 f32 per kb. lane l<16: M=l, K=kb*4+{0,1}; l>=16: K=kb*4+{2,3}
    v2f A[8];
    #pragma unroll
    for (int kb = 0; kb < 8; ++kb)
      A[kb] = *(const v2f*)(relRow + kb * 4 + half * 2);

    float maskv[8];
    #pragma unroll
    for (int r = 0; r < 8; ++r)
      maskv[r] = mask[(size_t)i * NN + j0 + r + 8 * half];
    const float dinv = 1.0f / deg[i];

    float dotAcc[8];
    #pragma unroll
    for (int r = 0; r < 8; ++r) dotAcc[r] = 0.f;

    for (int nb = 0; nb < DE / 16; ++nb) {
      const int n = nb * 16 + ln;
      v8f c = {0.f, 0.f, 0.f, 0.f, 0.f, 0.f, 0.f, 0.f};
      #pragma unroll
      for (int kb = 0; kb < 8; ++kb) {
        v2f bt = *(const v2f*)&ldsWB[(((kb * 18) + nb) * 32 + lane) * 2];
        c = __builtin_amdgcn_wmma_f32_16x16x4_f32(
            false, A[kb], false, bt, (short)0, c, false, false);
      }
      const float s1v = S1[(size_t)i * DE + n];
      const float b1v = ldsB1[n];
      const float w2v = ldsW2[n];
      #pragma unroll
      for (int r = 0; r < 8; ++r) {
        const int m = r + 8 * half;                 // edge row in tile
        float pre = maskv[r] * (s1v + ldsS2[m * DE + n] + c[r]) + b1v;
        dotAcc[r] += leaky(pre) * w2v;
      }
    }

    #pragma unroll
    for (int r = 0; r < 8; ++r) {
      float d = dotAcc[r];
      d += __shfl_xor(d, 1, 16);
      d += __shfl_xor(d, 2, 16);
      d += __shfl_xor(d, 4, 16);
      d += __shfl_xor(d, 8, 16);
      if (ln == r) {                                // owner of m = r + 8*half
        float w2s = leaky(d + b2v);
        sacc += w2s * maskv[r] * dinv;
      }
    }
  }

  if (ln < 8) {
    const int p = blockIdx.y * 4 + wave;            // 0..31
    const int j = j0 + ln + 8 * half;
    partials[p * NN + j] = sacc;
  }
}

// ---------------------------------------------------------------------------
// s[j] = sum_p partials[p,j];  seqNext[j,u] = s[j] * seqPrev[j,u]
// ---------------------------------------------------------------------------
__global__ __launch_bounds__(256) void scale_kernel(
    const float* __restrict__ partials, const float* __restrict__ seqPrev,
    float* __restrict__ seqNext)
{
  int idx = blockIdx.x * 256 + threadIdx.x;          // 512*128
  int j = idx >> 7;
  float s = 0.f;
  #pragma unroll 8
  for (int p = 0; p < 32; ++p) s += partials[p * NN + j];
  seqNext[idx] = s * seqPrev[idx];
}

// ---------------------------------------------------------------------------
// FC head: c=[seq0,seqF] (256); h1=leaky(c@Wf1+b); h2=leaky(h1@Wf2+b);
// out = leaky(h2@Wf3+b).  One block per row.
// ---------------------------------------------------------------------------
__global__ __launch_bounds__(256) void head_kernel(
    const float* __restrict__ seq0, const float* __restrict__ seqF,
    const float* __restrict__ Wf1, const float* __restrict__ bf1,
    const float* __restrict__ Wf2, const float* __restrict__ bf2,
    const float* __restrict__ Wf3, const float* __restrict__ bf3,
    float* __restrict__ out)
{
  __shared__ float c[256];
  __shared__ float h[256];
  __shared__ float red[256];
  int n = blockIdx.x, t = threadIdx.x;
  c[t] = (t < DU) ? seq0[n * DU + t] : seqF[n * DU + (t - DU)];
  __syncthreads();
  float a = bf1[t];
  #pragma unroll 4
  for (int u = 0; u < 256; ++u) a += c[u] * Wf1[u * 256 + t];
  a = leaky(a);
  h[t] = a; __syncthreads();
  float a2 = bf2[t];
  #pragma unroll 4
  for (int u = 0; u < 256; ++u) a2 += h[u] * Wf2[u * 256 + t];
  a2 = leaky(a2);
  red[t] = a2 * Wf3[t];
  __syncthreads();
  for (int s = 128; s > 0; s >>= 1) { if (t < s) red[t] += red[t + s]; __syncthreads(); }
  if (t == 0) out[n] = leaky(red[0] + bf3[0]);
}

// ---------------------------------------------------------------------------
extern "C" void kernel_launch(void* const* d_in, const int* in_sizes, int n_in,
                              void* d_out, int out_size, void* d_ws, size_t ws_size,
                              hipStream_t stream)
{
  const float* input = (const float*)d_in[0];
  const float* rel   = (const float*)d_in[1];
  const float* W_seq = (const float*)d_in[2];
  const float* b_seq = (const float*)d_in[3];
  const float* W_h11 = (const float*)d_in[4];
  const float* b_h11 = (const float*)d_in[5];
  const float* W_h12 = (const float*)d_in[6];
  const float* b_h12 = (const float*)d_in[7];
  const float* W_h21 = (const float*)d_in[8];
  const float* b_h21 = (const float*)d_in[9];
  const float* W_h22 = (const float*)d_in[10];
  const float* b_h22 = (const float*)d_in[11];
  const float* W_f1  = (const float*)d_in[12];
  const float* b_f1  = (const float*)d_in[13];
  const float* W_f2  = (const float*)d_in[14];
  const float* b_f2  = (const float*)d_in[15];
  const float* W_f3  = (const float*)d_in[16];
  const float* b_f3  = (const float*)d_in[17];

  float* ws = (float*)d_ws;
  float* seq0     = ws; ws += NN * DU;
  float* seq1     = ws; ws += NN * DU;
  float* seq2     = ws; ws += NN * DU;
  float* maskb    = ws; ws += NN * NN;
  float* degb     = ws; ws += NN;
  float* S1       = ws; ws += NN * DE;
  float* S2       = ws; ws += NN * DE;
  float* partials = ws; ws += 32 * NN;

  seq_embed_kernel<<<NN * DU / 256, 256, 0, stream>>>(input, W_seq, b_seq, seq0);
  mask_deg_kernel<<<NN, 128, 0, stream>>>(rel, maskb, degb);

  // hop 1
  s12_kernel<<<(NN * DE + 255) / 256, 256, 0, stream>>>(seq0, W_h11, S1, S2);
  edge_hop_kernel<<<dim3(NN / 16, 8), 128, 0, stream>>>(
      rel, W_h11 + (size_t)2 * DU * DE, b_h11, W_h12, b_h12,
      S1, S2, maskb, degb, partials);
  scale_kernel<<<NN * DU / 256, 256, 0, stream>>>(partials, seq0, seq1);

  // hop 2
  s12_kernel<<<(NN * DE + 255) / 256, 256, 0, stream>>>(seq1, W_h21, S1, S2);
  edge_hop_kernel<<<dim3(NN / 16, 8), 128, 0, stream>>>(
      rel, W_h21 + (size_t)2 * DU * DE, b_h21, W_h22, b_h22,
      S1, S2, maskb, degb, partials);
  scale_kernel<<<NN * DU / 256, 256, 0, stream>>>(partials, seq1, seq2);

  head_kernel<<<NN, 256, 0, stream>>>(seq0, seq2, W_f1, b_f1, W_f2, b_f2,
                                      W_f3, b_f3, (float*)d_out);
}